// BotImpact_19602230739467
// MI455X (gfx1250) — compile-verified
//
#include <hip/hip_runtime.h>

// ---------------------------------------------------------------------------
// BotImpact GAT pipeline for gfx1250 (MI455X).
//   xw = X @ W          -> fp32 WMMA (v_wmma_f32_16x16x4_f32), exact fp32
//   attention softmax   -> 3 atomic edge passes (ordered-uint max, add, agg)
//   heads               -> shuffle-reduce / LDS scalar kernels
// ---------------------------------------------------------------------------

typedef float v2f __attribute__((ext_vector_type(2)));
typedef float v8f __attribute__((ext_vector_type(8)));

#define FDIM 128   // IN == HC == 128
#define CDIM 64    // channels per head
#define NHEAD 2

__device__ __forceinline__ unsigned f2ord(float f) {
  unsigned u = __float_as_uint(f);
  return (u & 0x80000000u) ? ~u : (u | 0x80000000u);
}
__device__ __forceinline__ float ord2f(unsigned u) {
  return (u & 0x80000000u) ? __uint_as_float(u & 0x7FFFFFFFu)
                           : __uint_as_float(~u);
}
__device__ __forceinline__ float leakyf(float v, float s) {
  return v > 0.f ? v : s * v;
}

// ---------------------------------------------------------------------------
// GEMM: C[N x 128] = A[N x 128] * B[128 x 128], fp32 WMMA 16x16x4.
// Block = 256 threads (8 waves). Block computes 16 rows x 128 cols.
// Wave w computes cols [16w, 16w+16). A tile staged in LDS (8 KB).
// ---------------------------------------------------------------------------
__global__ void __launch_bounds__(256)
gemm_wmma_f32(const float* __restrict__ A, const float* __restrict__ B,
              float* __restrict__ C, int N) {
  __shared__ float As[16 * FDIM];

  const int block_row = blockIdx.x * 16;
  const int tid = threadIdx.x;

  // Stage A tile: 2048 floats, 8 per thread, coalesced. Clamp row for tail.
  #pragma unroll
  for (int i = 0; i < 8; ++i) {
    int idx = tid + i * 256;
    int r = idx >> 7;
    int c = idx & (FDIM - 1);
    int row = block_row + r;
    if (row >= N) row = N - 1;           // duplicate-row clamp (harmless)
    As[idx] = A[(size_t)row * FDIM + c];
  }
  __syncthreads();

  const int wave = tid >> 5;             // 0..7
  const int lane = tid & 31;
  const int col0 = wave * 16;
  const int m = lane & 15;               // row within A-frag / col within B-frag
  const int khalf = (lane >> 4) * 2;     // 0 or 2

  v8f acc = {};
  for (int k = 0; k < FDIM; k += 4) {
    v2f a, b;
    a.x = As[m * FDIM + k + khalf];
    a.y = As[m * FDIM + k + khalf + 1];
    b.x = B[(size_t)(k + khalf) * FDIM + col0 + m];
    b.y = B[(size_t)(k + khalf + 1) * FDIM + col0 + m];
    acc = __builtin_amdgcn_wmma_f32_16x16x4_f32(
        /*neg_a=*/false, a, /*neg_b=*/false, b,
        /*c_mod=*/(short)0, acc, /*reuse_a=*/false, /*reuse_b=*/false);
  }

  // C/D layout: VGPR r, lanes 0-15 -> M=r, lanes 16-31 -> M=8+r; N=lane&15.
  const int row_base = block_row + ((lane >> 4) << 3);
  const int col = col0 + m;
  #pragma unroll
  for (int r = 0; r < 8; ++r) {
    int row = row_base + r;
    if (row < N) C[(size_t)row * FDIM + col] = acc[r];
  }
}

// ---------------------------------------------------------------------------
// Per-node attention logits: a_src[n][h] = sum_c xw[n,h,c]*as[h,c]  (same dst)
// One wave per node; 32-lane shuffle reduction.
// ---------------------------------------------------------------------------
__global__ void att_scores(const float* __restrict__ xw,
                           const float* __restrict__ att_s,
                           const float* __restrict__ att_d,
                           float* __restrict__ a_src,
                           float* __restrict__ a_dst, int n) {
  int gt = blockIdx.x * blockDim.x + threadIdx.x;
  int node = gt >> 5;
  int lane = gt & 31;
  if (node >= n) return;
  #pragma unroll
  for (int h = 0; h < NHEAD; ++h) {
    size_t base = (size_t)node * FDIM + h * CDIM;
    int wb = h * CDIM;
    float s = xw[base + lane] * att_s[wb + lane] +
              xw[base + lane + 32] * att_s[wb + lane + 32];
    float d = xw[base + lane] * att_d[wb + lane] +
              xw[base + lane + 32] * att_d[wb + lane + 32];
    #pragma unroll
    for (int off = 16; off > 0; off >>= 1) {
      s += __shfl_down(s, off, 32);
      d += __shfl_down(d, off, 32);
    }
    if (lane == 0) {
      a_src[node * NHEAD + h] = s;
      a_dst[node * NHEAD + h] = d;
    }
  }
}

// ---------------------------------------------------------------------------
// Init: emax (ordered-uint) = 0 (< mapped -inf), denom = 0, agg = 0.
// ---------------------------------------------------------------------------
__global__ void init_conv(unsigned* __restrict__ emax, float* __restrict__ denom,
                          float* __restrict__ agg, int n) {
  size_t t = (size_t)blockIdx.x * blockDim.x + threadIdx.x;
  if (t < (size_t)n * NHEAD) { emax[t] = 0u; denom[t] = 0.f; }
  if (t < (size_t)n * FDIM) agg[t] = 0.f;
}

// ---------------------------------------------------------------------------
// Edge pass 1: segment max of leaky_relu(a_src[src]+a_dst[dst], 0.2) per dst.
// Thread per (edge, head). Edges e>=E are self loops.
// ---------------------------------------------------------------------------
__global__ void edge_max(const int* __restrict__ ei, int E, int n,
                         const float* __restrict__ a_src,
                         const float* __restrict__ a_dst,
                         unsigned* __restrict__ emax) {
  size_t t = (size_t)blockIdx.x * blockDim.x + threadIdx.x;
  if (t >= (size_t)(E + n) * NHEAD) return;
  int e = (int)(t >> 1), h = (int)(t & 1);
  int s, d;
  if (e < E) { s = ei[e]; d = ei[E + e]; } else { s = d = e - E; }
  float sc = leakyf(a_src[s * NHEAD + h] + a_dst[d * NHEAD + h], 0.2f);
  atomicMax(&emax[d * NHEAD + h], f2ord(sc));
}

// ---------------------------------------------------------------------------
// Edge pass 2: p = exp(score - emax[dst]); denom[dst] += p.
// ---------------------------------------------------------------------------
__global__ void edge_exp(const int* __restrict__ ei, int E, int n,
                         const float* __restrict__ a_src,
                         const float* __restrict__ a_dst,
                         const unsigned* __restrict__ emax,
                         float* __restrict__ p, float* __restrict__ denom) {
  size_t t = (size_t)blockIdx.x * blockDim.x + threadIdx.x;
  if (t >= (size_t)(E + n) * NHEAD) return;
  int e = (int)(t >> 1), h = (int)(t & 1);
  int s, d;
  if (e < E) { s = ei[e]; d = ei[E + e]; } else { s = d = e - E; }
  float sc = leakyf(a_src[s * NHEAD + h] + a_dst[d * NHEAD + h], 0.2f);
  float ex = expf(sc - ord2f(emax[d * NHEAD + h]));
  p[(size_t)e * NHEAD + h] = ex;
  atomicAdd(&denom[d * NHEAD + h], ex);
}

// ---------------------------------------------------------------------------
// Edge pass 3: out[dst,:] += xw[src,:] * (p/denom[dst]).  One wave per edge:
// each lane does 4 coalesced loads + 4 float atomic adds.
// ---------------------------------------------------------------------------
__global__ void edge_agg(const int* __restrict__ ei, int E, int n,
                         const float* __restrict__ xw,
                         const float* __restrict__ p,
                         const float* __restrict__ denom,
                         float* __restrict__ out) {
  size_t gt = (size_t)blockIdx.x * blockDim.x + threadIdx.x;
  int e = (int)(gt >> 5);
  int lane = (int)(gt & 31);
  if (e >= E + n) return;
  int s, d;
  if (e < E) { s = ei[e]; d = ei[E + e]; } else { s = d = e - E; }
  float w0 = p[(size_t)e * NHEAD + 0] / denom[d * NHEAD + 0];
  float w1 = p[(size_t)e * NHEAD + 1] / denom[d * NHEAD + 1];
  size_t sb = (size_t)s * FDIM;
  size_t db = (size_t)d * FDIM;
  #pragma unroll
  for (int i = 0; i < 4; ++i) {
    int c = lane + 32 * i;
    float w = (c < CDIM) ? w0 : w1;
    atomicAdd(&out[db + c], xw[sb + c] * w);
  }
}

// ---------------------------------------------------------------------------
// Bias (+ optional ReLU), in place.
// ---------------------------------------------------------------------------
__global__ void bias_act(float* __restrict__ out, const float* __restrict__ b,
                         int n, int do_relu) {
  size_t t = (size_t)blockIdx.x * blockDim.x + threadIdx.x;
  if (t >= (size_t)n * FDIM) return;
  float v = out[t] + b[t & (FDIM - 1)];
  if (do_relu) v = v > 0.f ? v : 0.f;
  out[t] = v;
}

// ---------------------------------------------------------------------------
// y heads: wave per (which, i). which: 0=y1, 1=yc0, 2=y0, 3=yc1.
// ---------------------------------------------------------------------------
__global__ void y_heads(const float* __restrict__ xZ2,
                        const float* __restrict__ xfZ2,
                        const int* __restrict__ treat,
                        const int* __restrict__ control,
                        const float* __restrict__ Wy1,
                        const float* __restrict__ by1,
                        const float* __restrict__ Wy0,
                        const float* __restrict__ by0,
                        float* __restrict__ out, int nt) {
  size_t gt = (size_t)blockIdx.x * blockDim.x + threadIdx.x;
  int job = (int)(gt >> 5);
  int lane = (int)(gt & 31);
  if (job >= 4 * nt) return;
  int which = job / nt;
  int i = job - which * nt;

  const float* feat;
  const float* W;
  float bias;
  int idx;
  switch (which) {
    case 0:  idx = treat[i];   feat = xZ2;  W = Wy1; bias = by1[0]; break;
    case 1:  idx = treat[i];   feat = xfZ2; W = Wy0; bias = by0[0]; break;
    case 2:  idx = control[i]; feat = xZ2;  W = Wy0; bias = by0[0]; break;
    default: idx = control[i]; feat = xfZ2; W = Wy1; bias = by1[0]; break;
  }
  size_t base = (size_t)idx * FDIM;
  float s = feat[base + lane] * W[lane] +
            feat[base + lane + 32] * W[lane + 32] +
            feat[base + lane + 64] * W[lane + 64] +
            feat[base + lane + 96] * W[lane + 96];
  #pragma unroll
  for (int off = 16; off > 0; off >>= 1) s += __shfl_down(s, off, 32);
  if (lane == 0) out[(size_t)which * nt + i] = leakyf(s + bias, 0.01f);
}

// ---------------------------------------------------------------------------
// Propensity head: tprob = leaky(leaky(x@Wp1+bp1)@Wp2+bp2). Block per node.
// ---------------------------------------------------------------------------
__global__ void __launch_bounds__(64)
propen_head(const float* __restrict__ feat, const float* __restrict__ Wp1,
            const float* __restrict__ bp1, const float* __restrict__ Wp2,
            const float* __restrict__ bp2, float* __restrict__ out, int n) {
  __shared__ float xr[FDIM];
  __shared__ float hid[CDIM];
  int node = blockIdx.x;
  int t = threadIdx.x;  // 64 threads
  if (node >= n) return;
  size_t base = (size_t)node * FDIM;
  xr[t] = feat[base + t];
  xr[t + 64] = feat[base + t + 64];
  __syncthreads();
  float acc = bp1[t];
  #pragma unroll 8
  for (int k = 0; k < FDIM; ++k) acc += xr[k] * Wp1[k * CDIM + t];
  hid[t] = leakyf(acc, 0.01f);
  __syncthreads();
  if (t < 2) {
    float a2 = bp2[t];
    #pragma unroll 8
    for (int k = 0; k < CDIM; ++k) a2 += hid[k] * Wp2[k * 2 + t];
    out[(size_t)node * 2 + t] = leakyf(a2, 0.01f);
  }
}

// ---------------------------------------------------------------------------
// Host orchestration
// ---------------------------------------------------------------------------
static inline int cdiv(long long a, long long b) { return (int)((a + b - 1) / b); }

static void run_gat(const float* X, const float* W, const float* att_s,
                    const float* att_d, const float* b, const int* ei,
                    int N, int E,
                    float* buf_xw, float* a_src, float* a_dst, unsigned* emax,
                    float* denom, float* p, float* out, int do_relu,
                    hipStream_t stream) {
  long long NE = (long long)N * FDIM;
  long long ETH = (long long)(E + N) * NHEAD;      // edge-pass threads
  long long EAG = (long long)(E + N) * 32;         // wave per edge

  gemm_wmma_f32<<<cdiv(N, 16), 256, 0, stream>>>(X, W, buf_xw, N);
  att_scores<<<cdiv((long long)N * 32, 256), 256, 0, stream>>>(
      buf_xw, att_s, att_d, a_src, a_dst, N);
  init_conv<<<cdiv(NE, 256), 256, 0, stream>>>(emax, denom, out, N);
  edge_max<<<cdiv(ETH, 256), 256, 0, stream>>>(ei, E, N, a_src, a_dst, emax);
  edge_exp<<<cdiv(ETH, 256), 256, 0, stream>>>(ei, E, N, a_src, a_dst, emax,
                                               p, denom);
  edge_agg<<<cdiv(EAG, 256), 256, 0, stream>>>(ei, E, N, buf_xw, p, denom, out);
  bias_act<<<cdiv(NE, 256), 256, 0, stream>>>(out, b, N, do_relu);
}

extern "C" void kernel_launch(void* const* d_in, const int* in_sizes, int n_in,
                              void* d_out, int out_size, void* d_ws,
                              size_t ws_size, hipStream_t stream) {
  const float* x       = (const float*)d_in[0];
  const int*   ei      = (const int*)d_in[1];
  const float* fx      = (const float*)d_in[2];
  const int*   fei     = (const int*)d_in[3];
  const int*   treat   = (const int*)d_in[4];
  const int*   control = (const int*)d_in[5];
  const float* W1  = (const float*)d_in[6];
  const float* as1 = (const float*)d_in[7];
  const float* ad1 = (const float*)d_in[8];
  const float* b1  = (const float*)d_in[9];
  const float* W2  = (const float*)d_in[10];
  const float* as2 = (const float*)d_in[11];
  const float* ad2 = (const float*)d_in[12];
  const float* b2  = (const float*)d_in[13];
  const float* Wy1 = (const float*)d_in[14];
  const float* by1 = (const float*)d_in[15];
  const float* Wy0 = (const float*)d_in[16];
  const float* by0 = (const float*)d_in[17];
  const float* Wp1 = (const float*)d_in[18];
  const float* bp1 = (const float*)d_in[19];
  const float* Wp2 = (const float*)d_in[20];
  const float* bp2 = (const float*)d_in[21];

  const int N  = in_sizes[0] / FDIM;   // 100000
  const int E  = in_sizes[1] / 2;      // 1600000
  const int NT = in_sizes[4];          // 20000

  // Output layout (return order): y1, yc0, y0, yc1, xZ2, xfZ2, tprob, tprob_f
  float* out    = (float*)d_out;
  float* y_out  = out;
  float* xZ2    = out + (size_t)4 * NT;
  float* xfZ2   = xZ2 + (size_t)N * FDIM;
  float* tprob  = xfZ2 + (size_t)N * FDIM;
  float* tprobf = tprob + (size_t)N * 2;

  // Workspace layout
  float* ws       = (float*)d_ws;
  float* buf_xw   = ws;                                   // N*128
  float* buf_h    = buf_xw + (size_t)N * FDIM;            // N*128
  float* a_src    = buf_h + (size_t)N * FDIM;             // N*2
  float* a_dst    = a_src + (size_t)N * NHEAD;            // N*2
  unsigned* emax  = (unsigned*)(a_dst + (size_t)N * NHEAD);   // N*2
  float* denom    = (float*)(emax + (size_t)N * NHEAD);   // N*2
  float* p        = denom + (size_t)N * NHEAD;            // (E+N)*2

  // Real graph: layer 1 (ReLU) -> buf_h; layer 2 -> xZ2 (no ReLU)
  run_gat(x, W1, as1, ad1, b1, ei, N, E, buf_xw, a_src, a_dst, emax, denom, p,
          buf_h, 1, stream);
  run_gat(buf_h, W2, as2, ad2, b2, ei, N, E, buf_xw, a_src, a_dst, emax, denom,
          p, xZ2, 0, stream);

  // Fake graph
  run_gat(fx, W1, as1, ad1, b1, fei, N, E, buf_xw, a_src, a_dst, emax, denom, p,
          buf_h, 1, stream);
  run_gat(buf_h, W2, as2, ad2, b2, fei, N, E, buf_xw, a_src, a_dst, emax, denom,
          p, xfZ2, 0, stream);

  // Heads
  y_heads<<<cdiv((long long)4 * NT * 32, 256), 256, 0, stream>>>(
      xZ2, xfZ2, treat, control, Wy1, by1, Wy0, by0, y_out, NT);
  propen_head<<<N, 64, 0, stream>>>(xZ2, Wp1, bp1, Wp2, bp2, tprob, N);
  propen_head<<<N, 64, 0, stream>>>(xfZ2, Wp1, bp1, Wp2, bp2, tprobf, N);
}